// GAT_39092792328213
// MI455X (gfx1250) — compile-verified
//
#include <hip/hip_runtime.h>
#include <math.h>

typedef float v2f __attribute__((ext_vector_type(2)));
typedef float v8f __attribute__((ext_vector_type(8)));

#define NNODES   50000
#define NEDGE    800000
#define ETOT     (NEDGE + NNODES)
#define NEG_SLOPE 0.2f
#define GAT_EPS   1e-16f

// Native CDNA5 float atomics (GLOBAL_ATOMIC_ADD_F32 / GLOBAL_ATOMIC_MAX_NUM_F32),
// avoiding CAS-loop lowering of the safe variants. Accumulators are L2-resident.
__device__ __forceinline__ void atomAddF32(float* p, float v) {
  unsafeAtomicAdd(p, v);
}
__device__ __forceinline__ void atomMaxF32(float* p, float v) {
  unsafeAtomicMax(p, v);
}

// ---------------------------------------------------------------------------
// fp32 WMMA GEMM:  C[M,N] = A[M,K] @ B[K,N] (+ bias), via V_WMMA_F32_16X16X4_F32
// Block = 256 threads = 8 waves; each wave owns a 16-row strip, loops 16-col tiles.
// A 16x4 per-lane layout: v2f at A[row(lane&15)][k0 + 2*(lane>>4)]
// B 4x16 per-lane layout: rows striped across lanes (K=k0+2*half in .x, +1 in .y)
// D v8f: element r -> row = row0 + r + 8*(lane>>4), col = n0 + (lane&15)
// ---------------------------------------------------------------------------
template<int K, int N>
__global__ __launch_bounds__(256) void gemm_bias_wmma(
    const float* __restrict__ A, const float* __restrict__ B,
    const float* __restrict__ bias, float* __restrict__ C, int M)
{
  const int lane = threadIdx.x & 31;
  const int wave = threadIdx.x >> 5;
  const int half = lane >> 4;
  const int l16  = lane & 15;
  const int row0 = blockIdx.x * 128 + wave * 16;
  int mload = row0 + l16;
  if (mload >= M) mload = M - 1;               // clamp OOB loads; stores are guarded
  const float* Arow = A + (size_t)mload * K;

  for (int t = 0; t < N / 16; ++t) {
    const int n0 = t * 16;
    v8f acc = {};
#pragma unroll
    for (int k0 = 0; k0 < K; k0 += 4) {
      const int ka = k0 + 2 * half;
      v2f a = *(const v2f*)(Arow + ka);
      v2f b;
      b.x = B[(size_t)ka * N + n0 + l16];
      b.y = B[(size_t)(ka + 1) * N + n0 + l16];
      acc = __builtin_amdgcn_wmma_f32_16x16x4_f32(
          /*neg_a=*/false, a, /*neg_b=*/false, b,
          /*c_mod=*/(short)0, acc, /*reuse_a=*/false, /*reuse_b=*/false);
    }
    const float bv = bias ? bias[n0 + l16] : 0.0f;
#pragma unroll
    for (int r = 0; r < 8; ++r) {
      const int mr = row0 + r + 8 * half;
      if (mr < M) C[(size_t)mr * N + n0 + l16] = acc[r] + bv;
    }
  }
}

// ---------------------------------------------------------------------------
// Wave-per-node: attention dot products a_src/a_dst, plus per-call state init
// (amax=-inf, denom=0, agg row zeroed) so replays are deterministic.
// ---------------------------------------------------------------------------
template<int H, int C>
__global__ __launch_bounds__(256) void att_init_kernel(
    const float* __restrict__ xw, const float* __restrict__ att_src,
    const float* __restrict__ att_dst, float* __restrict__ a_src,
    float* __restrict__ a_dst, float* __restrict__ amax,
    float* __restrict__ denom, float* __restrict__ agg, int n)
{
  const int lane = threadIdx.x & 31;
  const int node = (int)((blockIdx.x * (unsigned)blockDim.x + threadIdx.x) >> 5);
  if (node >= n) return;
  const float* row = xw + (size_t)node * (H * C);
#pragma unroll
  for (int h = 0; h < H; ++h) {
    float ssum = 0.f, dsum = 0.f;
#pragma unroll
    for (int i = 0; i < C / 32; ++i) {
      const int c = i * 32 + lane;
      const float v = row[h * C + c];
      ssum += v * att_src[h * C + c];
      dsum += v * att_dst[h * C + c];
    }
#pragma unroll
    for (int off = 16; off > 0; off >>= 1) {
      ssum += __shfl_xor(ssum, off, 32);
      dsum += __shfl_xor(dsum, off, 32);
    }
    if (lane == 0) {
      a_src[node * H + h] = ssum;
      a_dst[node * H + h] = dsum;
      amax[node * H + h]  = -3.0e38f;
      denom[node * H + h] = 0.f;
    }
  }
  float* arow = agg + (size_t)node * (H * C);
#pragma unroll
  for (int i = 0; i < (H * C) / 32; ++i) arow[i * 32 + lane] = 0.f;
}

__device__ __forceinline__ void edge_sd(const int* ei, int e, int& s, int& d) {
  if (e < NEDGE) { s = ei[e]; d = ei[NEDGE + e]; }
  else           { s = e - NEDGE; d = s; }       // implicit self-loops
}

// Pass A: segment max via L2-resident native float atomic max
template<int H>
__global__ void edge_max_kernel(const int* __restrict__ ei,
    const float* __restrict__ a_src, const float* __restrict__ a_dst,
    float* __restrict__ amax)
{
  const int e = blockIdx.x * blockDim.x + threadIdx.x;
  if (e >= ETOT) return;
  int s, d; edge_sd(ei, e, s, d);
#pragma unroll
  for (int h = 0; h < H; ++h) {
    const float a = a_src[s * H + h] + a_dst[d * H + h];
    const float alpha = a > 0.f ? a : NEG_SLOPE * a;
    atomMaxF32(&amax[d * H + h], alpha);
  }
}

// Pass B: ex = exp(alpha - amax[dst]); cache per-edge, segment-sum via atomic add
template<int H>
__global__ void edge_exp_kernel(const int* __restrict__ ei,
    const float* __restrict__ a_src, const float* __restrict__ a_dst,
    const float* __restrict__ amax, float* __restrict__ ex,
    float* __restrict__ denom)
{
  const int e = blockIdx.x * blockDim.x + threadIdx.x;
  if (e >= ETOT) return;
  int s, d; edge_sd(ei, e, s, d);
#pragma unroll
  for (int h = 0; h < H; ++h) {
    const float a = a_src[s * H + h] + a_dst[d * H + h];
    const float alpha = a > 0.f ? a : NEG_SLOPE * a;
    const float v = expf(alpha - amax[d * H + h]);
    ex[e * H + h] = v;
    atomAddF32(&denom[d * H + h], v);
  }
}

// Pass C: wave-per-edge message scatter. Coalesced gather of xw[src] row,
// coalesced native f32 atomic adds into agg[dst] row (L2-resident accumulators).
template<int H, int C>
__global__ __launch_bounds__(256) void edge_aggr_kernel(const int* __restrict__ ei,
    const float* __restrict__ ex, const float* __restrict__ denom,
    const float* __restrict__ xw, float* __restrict__ agg)
{
  const int lane = threadIdx.x & 31;
  const int e = (int)(((size_t)blockIdx.x * blockDim.x + threadIdx.x) >> 5);
  if (e >= ETOT) return;
  int s, d; edge_sd(ei, e, s, d);
  float coef[H];
#pragma unroll
  for (int h = 0; h < H; ++h)
    coef[h] = ex[e * H + h] / (denom[d * H + h] + GAT_EPS);
  const float* srow = xw + (size_t)s * (H * C);
  float* drow = agg + (size_t)d * (H * C);
#pragma unroll
  for (int i = 0; i < (H * C) / 32; ++i) {
    const int idx = i * 32 + lane;
    const int h = idx / C;   // compile-time per i
    atomAddF32(&drow[idx], coef[h] * srow[idx]);
  }
}

// Finalize layer 1: h = elu(agg + bias), in place
__global__ void bias_elu_kernel(float* __restrict__ agg,
    const float* __restrict__ bias, int total, int mask)
{
  const int i = blockIdx.x * blockDim.x + threadIdx.x;
  if (i >= total) return;
  const float v = agg[i] + bias[i & mask];
  agg[i] = v > 0.f ? v : expm1f(v);
}

// Finalize layer 2: out = agg + bias
__global__ void bias_out_kernel(const float* __restrict__ agg,
    const float* __restrict__ bias, float* __restrict__ out, int total, int mask)
{
  const int i = blockIdx.x * blockDim.x + threadIdx.x;
  if (i >= total) return;
  out[i] = agg[i] + bias[i & mask];
}

extern "C" void kernel_launch(void* const* d_in, const int* in_sizes, int n_in,
                              void* d_out, int out_size, void* d_ws, size_t ws_size,
                              hipStream_t stream) {
  const float* x        = (const float*)d_in[0];
  const int*   ei       = (const int*)d_in[1];     // [2, 800000]
  const float* W_emb    = (const float*)d_in[2];
  const float* b_emb    = (const float*)d_in[3];
  const float* W1       = (const float*)d_in[4];
  const float* att_src1 = (const float*)d_in[5];
  const float* att_dst1 = (const float*)d_in[6];
  const float* b1       = (const float*)d_in[7];
  const float* W2       = (const float*)d_in[8];
  const float* att_src2 = (const float*)d_in[9];
  const float* att_dst2 = (const float*)d_in[10];
  const float* b2       = (const float*)d_in[11];
  float* out = (float*)d_out;

  // ---- workspace layout (floats) ----
  float* ws = (float*)d_ws;
  float* xw1   = ws;                          // 50000*256
  float* agg1  = xw1  + (size_t)NNODES * 256; // 50000*256 (agg1 -> h1 in place)
  float* h0    = agg1 + (size_t)NNODES * 256; // 50000*128 (h0, reused as xw2)
  float* as1   = h0   + (size_t)NNODES * 128; // N*2
  float* ad1   = as1  + NNODES * 2;
  float* amax1 = ad1  + NNODES * 2;
  float* den1  = amax1 + NNODES * 2;
  float* as2   = den1 + NNODES * 2;           // N*1
  float* ad2   = as2  + NNODES;
  float* amax2 = ad2  + NNODES;
  float* den2  = amax2 + NNODES;
  float* ex1   = den2 + NNODES;               // ETOT*2
  float* ex2   = ex1  + (size_t)ETOT * 2;     // ETOT
  float* agg2  = xw1;                         // reuse xw1 region after layer 1

  const int gemmBlocks = (NNODES + 127) / 128;
  const int nodeWaveBlocks = (NNODES * 32 + 255) / 256;
  const int edgeBlocks = (ETOT + 255) / 256;
  const int edgeWaveBlocks = (ETOT * 32 + 255) / 256;

  // node embedding: h0 = x @ W_emb + b_emb
  gemm_bias_wmma<128, 128><<<gemmBlocks, 256, 0, stream>>>(x, W_emb, b_emb, h0, NNODES);

  // ---- GAT layer 1 (H=2, C=128) ----
  gemm_bias_wmma<128, 256><<<gemmBlocks, 256, 0, stream>>>(h0, W1, nullptr, xw1, NNODES);
  att_init_kernel<2, 128><<<nodeWaveBlocks, 256, 0, stream>>>(
      xw1, att_src1, att_dst1, as1, ad1, amax1, den1, agg1, NNODES);
  edge_max_kernel<2><<<edgeBlocks, 256, 0, stream>>>(ei, as1, ad1, amax1);
  edge_exp_kernel<2><<<edgeBlocks, 256, 0, stream>>>(ei, as1, ad1, amax1, ex1, den1);
  edge_aggr_kernel<2, 128><<<edgeWaveBlocks, 256, 0, stream>>>(ei, ex1, den1, xw1, agg1);
  bias_elu_kernel<<<(NNODES * 256 + 255) / 256, 256, 0, stream>>>(agg1, b1, NNODES * 256, 255);

  // ---- GAT layer 2 (H=1, C=128) ----
  float* xw2 = h0;  // reuse
  gemm_bias_wmma<256, 128><<<gemmBlocks, 256, 0, stream>>>(agg1, W2, nullptr, xw2, NNODES);
  att_init_kernel<1, 128><<<nodeWaveBlocks, 256, 0, stream>>>(
      xw2, att_src2, att_dst2, as2, ad2, amax2, den2, agg2, NNODES);
  edge_max_kernel<1><<<edgeBlocks, 256, 0, stream>>>(ei, as2, ad2, amax2);
  edge_exp_kernel<1><<<edgeBlocks, 256, 0, stream>>>(ei, as2, ad2, amax2, ex2, den2);
  edge_aggr_kernel<1, 128><<<edgeWaveBlocks, 256, 0, stream>>>(ei, ex2, den2, xw2, agg2);
  bias_out_kernel<<<(NNODES * 128 + 255) / 256, 256, 0, stream>>>(agg2, b2, out, NNODES * 128, 127);

  (void)in_sizes; (void)n_in; (void)out_size; (void)ws_size;
}